// GNN_17738214933082
// MI455X (gfx1250) — compile-verified
//
#include <hip/hip_runtime.h>

#define N_NODES 100000
#define E_EDGES 1600000
#define DIM 128
#define LDS_STRIDE 132   // 128 + 4 pad: de-conflicts stride-128 A-frag LDS reads

typedef __attribute__((ext_vector_type(2))) float v2f;
typedef __attribute__((ext_vector_type(8))) float v8f;

// ---------------------------------------------------------------- zero fill
__global__ void GNN_zero_kernel(float* __restrict__ p, long n) {
  long i = (long)blockIdx.x * blockDim.x + threadIdx.x;
  long stride = (long)gridDim.x * blockDim.x;
  for (; i < n; i += stride) p[i] = 0.0f;
}

// ------------------------------------------------------------- degree count
__global__ void GNN_count_kernel(const long long* __restrict__ ei,
                                 float* __restrict__ cnt) {
  int i = blockIdx.x * blockDim.x + threadIdx.x;
  int stride = gridDim.x * blockDim.x;
  for (; i < E_EDGES; i += stride) {
    int dst = (int)ei[E_EDGES + i];
    atomicAdd(&cnt[dst], 1.0f);
  }
}

// ------------------------------------------------- edge scatter (sum aggregation)
// one wave32 per edge; lane handles 4 consecutive features (float4 gather,
// 4x global_atomic_add_f32 into the L2-resident accumulator table)
__global__ void GNN_scatter_kernel(const long long* __restrict__ ei,
                                   const float* __restrict__ feat,
                                   float* __restrict__ agg) {
  int lane = threadIdx.x & 31;
  int wave = blockIdx.x * (blockDim.x >> 5) + (threadIdx.x >> 5);
  int nwaves = gridDim.x * (blockDim.x >> 5);
  for (int e = wave; e < E_EDGES; e += nwaves) {
    int src = (int)ei[e];
    int dst = (int)ei[E_EDGES + e];
    const float4 v = *(const float4*)(feat + (long)src * DIM + lane * 4);
    float* a = agg + (long)dst * DIM + lane * 4;
    atomicAdd(a + 0, v.x);
    atomicAdd(a + 1, v.y);
    atomicAdd(a + 2, v.z);
    atomicAdd(a + 3, v.w);
  }
}

// --------------------------------------------------------- fused SAGE GEMM
// out[tile] = (agg * inv_cnt) @ Wl + x @ Wr + bias  (optional relu)
// block: 256 threads = 8 waves, 16 rows x 128 cols per block.
// wave w computes the 16x16 tile at columns [16w, 16w+16) using
// V_WMMA_F32_16X16X4_F32, 2 WMMAs per K-step (x-path + agg-path).
template <bool RELU>
__global__ __launch_bounds__(256) void GNN_sage_gemm_kernel(
    const float* __restrict__ xin,   // [N,128] self features
    const float* __restrict__ agg,   // [N,128] summed neighbor features
    const float* __restrict__ cnt,   // [N]     degree counts
    const float* __restrict__ Wl,    // [128,128] row-major (K x N)
    const float* __restrict__ Wr,    // [128,128]
    const float* __restrict__ bias,  // [128]
    float* __restrict__ out)         // [N,128]
{
  __shared__ float xs[16 * LDS_STRIDE];
  __shared__ float as[16 * LDS_STRIDE];

  const int tid = threadIdx.x;
  const int row0 = blockIdx.x * 16;  // N = 6250 * 16 exactly, no partial tiles

  // Stage the 16x128 input tiles into LDS; fold the mean (1/max(cnt,1))
  // into the agg tile here so the GEMM is a plain MAC chain.
  for (int e = tid; e < 16 * DIM; e += 256) {
    int r = e >> 7;
    int c = e & 127;
    long g = (long)(row0 + r) * DIM + c;
    float inv = 1.0f / fmaxf(cnt[row0 + r], 1.0f);
    xs[r * LDS_STRIDE + c] = xin[g];
    as[r * LDS_STRIDE + c] = agg[g] * inv;
  }
  __syncthreads();

  const int wave = tid >> 5;
  const int lane = tid & 31;
  const int half = lane >> 4;   // 0: K={kk,kk+1}, 1: K={kk+2,kk+3}
  const int l16  = lane & 15;
  const int col  = wave * 16 + l16;

  // C/D layout: VGPR j -> (M = j + 8*half, N = l16); bias is per-column.
  v8f acc;
  const float bv = bias[col];
#pragma unroll
  for (int j = 0; j < 8; ++j) acc[j] = bv;

  const int arow = l16;  // A-matrix row held by this lane

  for (int kk = 0; kk < DIM; kk += 4) {
    const int ks = kk + half * 2;

    // A fragments (16x4 fp32) from LDS
    v2f ax, aa;
    ax[0] = xs[arow * LDS_STRIDE + ks];
    ax[1] = xs[arow * LDS_STRIDE + ks + 1];
    aa[0] = as[arow * LDS_STRIDE + ks];
    aa[1] = as[arow * LDS_STRIDE + ks + 1];

    // B fragments (4x16 fp32) from global (weights are L2/WGP$-resident)
    v2f br, bl;
    br[0] = Wr[(long)ks * DIM + col];
    br[1] = Wr[(long)(ks + 1) * DIM + col];
    bl[0] = Wl[(long)ks * DIM + col];
    bl[1] = Wl[(long)(ks + 1) * DIM + col];

    // acc += x_tile @ Wr ; acc += agg_tile @ Wl
    acc = __builtin_amdgcn_wmma_f32_16x16x4_f32(
        false, ax, false, br, (short)0, acc, false, false);
    acc = __builtin_amdgcn_wmma_f32_16x16x4_f32(
        false, aa, false, bl, (short)0, acc, false, false);
  }

#pragma unroll
  for (int j = 0; j < 8; ++j) {
    const int m = j + half * 8;
    float v = acc[j];
    if (RELU) v = fmaxf(v, 0.0f);
    out[(long)(row0 + m) * DIM + col] = v;
  }
}

// ------------------------------------------------------------------ launch
extern "C" void kernel_launch(void* const* d_in, const int* in_sizes, int n_in,
                              void* d_out, int out_size, void* d_ws, size_t ws_size,
                              hipStream_t stream) {
  (void)in_sizes; (void)n_in; (void)out_size; (void)ws_size;

  const float*     x   = (const float*)d_in[0];
  const long long* ei  = (const long long*)d_in[1];  // int64 [2, E]
  const float*     W1l = (const float*)d_in[2];
  const float*     W1r = (const float*)d_in[3];
  const float*     b1  = (const float*)d_in[4];
  const float*     W2l = (const float*)d_in[5];
  const float*     W2r = (const float*)d_in[6];
  const float*     b2  = (const float*)d_in[7];
  float* out = (float*)d_out;

  float* cnt = (float*)d_ws;          // N floats
  float* agg = cnt + N_NODES;         // N*128 floats

  const int gemm_blocks = N_NODES / 16;  // 6250, exact

  // Layer 1
  GNN_zero_kernel<<<4096, 256, 0, stream>>>(cnt, (long)N_NODES * (1 + DIM));
  GNN_count_kernel<<<2048, 256, 0, stream>>>(ei, cnt);
  GNN_scatter_kernel<<<8192, 256, 0, stream>>>(ei, x, agg);
  GNN_sage_gemm_kernel<true><<<gemm_blocks, 256, 0, stream>>>(
      x, agg, cnt, W1l, W1r, b1, out);              // out now holds h

  // Layer 2 (degrees unchanged; reuse cnt)
  GNN_zero_kernel<<<4096, 256, 0, stream>>>(agg, (long)N_NODES * DIM);
  GNN_scatter_kernel<<<8192, 256, 0, stream>>>(ei, out, agg);
  GNN_sage_gemm_kernel<false><<<gemm_blocks, 256, 0, stream>>>(
      out, agg, cnt, W2l, W2r, b2, out);            // tile-disjoint in/out reuse
}